// Attention_56607668961538
// MI455X (gfx1250) — compile-verified
//
#include <hip/hip_runtime.h>
#include <hip/hip_bf16.h>

// MI455X (gfx1250) attention block, fp32 end-to-end via V_WMMA_F32_16X16X4_F32,
// with async global->LDS tile staging (ASYNCcnt) + double buffering in GEMMs.
// B=16, N=1024, C=1024, H=16, Dh=64.
// ws layout: q | k | v | o, each 16*16*1024*64 floats (64MB) -> 256MB total.

#define BATCH 16
#define SEQ   1024
#define CDIM  1024
#define HEADS 16
#define DH    64

typedef __attribute__((ext_vector_type(2))) float v2f;
typedef __attribute__((ext_vector_type(8))) float v8f;

__device__ __forceinline__ v8f wmma4(v2f a, v2f b, v8f c) {
    // D = A(16x4,f32) * B(4x16,f32) + C(16x16,f32)
    return __builtin_amdgcn_wmma_f32_16x16x4_f32(false, a, false, b, (short)0, c,
                                                 false, false);
}

// Low 32 bits of a flat shared-aperture pointer are the LDS byte address
// (ISA 10.2).
__device__ __forceinline__ unsigned lds_addr(const void* p) {
    return (unsigned)(unsigned long long)p;
}

// Async cache->LDS copies, tracked by ASYNCcnt (CDNA5 ISA).
__device__ __forceinline__ void async_copy_b128(unsigned lds, const float* g) {
    asm volatile("global_load_async_to_lds_b128 %0, %1, off"
                 :: "v"(lds), "v"(g) : "memory");
}
__device__ __forceinline__ void async_copy_b64(unsigned lds, const float* g) {
    asm volatile("global_load_async_to_lds_b64 %0, %1, off"
                 :: "v"(lds), "v"(g) : "memory");
}
__device__ __forceinline__ void wait_async0() {
    asm volatile("s_wait_asynccnt 0x0" ::: "memory");
}

// ---------------------------------------------------------------------------
// Issue one K-chunk stage: A tile 128x16 (2 x B128 per thread),
// B tile 32x16 (1 x B64 per thread). 256 threads.
// ---------------------------------------------------------------------------
__device__ __forceinline__ void issue_stage(
    const float* __restrict__ A, const float* __restrict__ B, int K,
    int m0, int n0, int k0, float* As, float* Bs, int tid)
{
    #pragma unroll
    for (int i = 0; i < 2; ++i) {
        const int flat = tid + i * 256;
        const int row  = flat >> 2;
        const int c4   = (flat & 3) << 2;
        async_copy_b128(lds_addr(&As[row * 16 + c4]),
                        &A[(size_t)(m0 + row) * K + k0 + c4]);
    }
    {
        const int row = tid >> 3;
        const int c2  = (tid & 7) << 1;
        async_copy_b64(lds_addr(&Bs[row * 16 + c2]),
                       &B[(size_t)(n0 + row) * K + k0 + c2]);
    }
}

// ---------------------------------------------------------------------------
// Shared GEMM mainloop: C[128x32 block] = A[M,K] * B[N,K]^T, row-major, K%16==0.
// 256 threads = 8 waves; wave (wm,wn) owns rows [wm*32,+32) x cols [wn*16,+16)
// as two 16x16 WMMA accumulators. Double-buffered async LDS staging:
//   issue loads for chunk k+1 -> compute chunk k -> s_wait_asynccnt 0 -> barrier
// ---------------------------------------------------------------------------
__device__ __forceinline__ void gemm_mainloop_nt(
    const float* __restrict__ A, const float* __restrict__ B, int K,
    int m0, int n0, float* As, float* Bs, v8f& c0, v8f& c1)
{
    const int tid  = threadIdx.x;
    const int lane = tid & 31;
    const int wv   = tid >> 5;
    const int wm   = wv >> 1;        // 0..3
    const int wn   = wv & 1;         // 0..1
    const int lh   = lane & 15;
    const int lg   = lane >> 4;      // 0/1
    const int ABUF = 128 * 16;
    const int BBUF = 32 * 16;

    // Prologue: stage chunk 0
    issue_stage(A, B, K, m0, n0, 0, As, Bs, tid);
    wait_async0();
    __syncthreads();

    int cur = 0;
    for (int k0 = 0; k0 < K; k0 += 16) {
        float* Ac = As + cur * ABUF;
        float* Bc = Bs + cur * BBUF;
        // Pipeline: kick off next chunk into the other buffer
        if (k0 + 16 < K)
            issue_stage(A, B, K, m0, n0, k0 + 16,
                        As + (cur ^ 1) * ABUF, Bs + (cur ^ 1) * BBUF, tid);

        #pragma unroll
        for (int kk = 0; kk < 16; kk += 4) {
            // A frag: lane lh holds row m=lh, k = kk + 2*lg + {0,1}
            v2f a0 = *(const v2f*)(&Ac[(wm * 32      + lh) * 16 + kk + 2 * lg]);
            v2f a1 = *(const v2f*)(&Ac[(wm * 32 + 16 + lh) * 16 + kk + 2 * lg]);
            // B frag: lane lh holds col n=lh, k = kk + 2*lg + {0,1}
            v2f b  = *(const v2f*)(&Bc[(wn * 16 + lh) * 16 + kk + 2 * lg]);
            c0 = wmma4(a0, b, c0);
            c1 = wmma4(a1, b, c1);
        }

        wait_async0();      // my async copies into the other buffer are done
        __syncthreads();    // everyone's copies done; everyone done reading cur
        cur ^= 1;
    }
}

// ---------------------------------------------------------------------------
// Kernel 1: qkv = x @ qkv_w^T, scattered into q/k/v buffers [b][h][n][d]
// ---------------------------------------------------------------------------
__global__ void __launch_bounds__(256)
qkv_gemm_kernel(const float* __restrict__ X, const float* __restrict__ W,
                float* __restrict__ qb, float* __restrict__ kb,
                float* __restrict__ vb)
{
    __shared__ float As[2 * 128 * 16];
    __shared__ float Bs[2 * 32 * 16];
    const int lane = threadIdx.x & 31;
    const int wv   = threadIdx.x >> 5;
    const int wm   = wv >> 1, wn = wv & 1;
    const int lh   = lane & 15, lg = lane >> 4;
    const int m0   = blockIdx.y * 128;
    const int n0   = blockIdx.x * 32;

    v8f c0 = {}, c1 = {};
    gemm_mainloop_nt(X, W, CDIM, m0, n0, As, Bs, c0, c1);

    const int n = n0 + wn * 16 + lh;       // 0..3071
    const int j = n >> 10;                 // 0=q 1=k 2=v
    const int h = (n >> 6) & (HEADS - 1);
    const int d = n & (DH - 1);
    float* dst = (j == 0) ? qb : ((j == 1) ? kb : vb);

    #pragma unroll
    for (int s2 = 0; s2 < 2; ++s2) {
        v8f c;
        if (s2) c = c1; else c = c0;
        #pragma unroll
        for (int v = 0; v < 8; ++v) {
            const int m  = m0 + wm * 32 + s2 * 16 + v + 8 * lg;   // b*N + n
            const int b  = m >> 10;
            const int nn = m & (SEQ - 1);
            dst[(((size_t)b * HEADS + h) * SEQ + nn) * DH + d] = c[v];
        }
    }
}

// ---------------------------------------------------------------------------
// Kernel 2: per-head RMSNorm on q and k, in place. One wave per 64-elem row.
// ---------------------------------------------------------------------------
__global__ void __launch_bounds__(256)
rmsnorm_kernel(float* __restrict__ qb, float* __restrict__ kb,
               const float* __restrict__ qw, const float* __restrict__ kw)
{
    const int  lane  = threadIdx.x & 31;
    const int  wv    = threadIdx.x >> 5;
    const long row   = (long)blockIdx.x * 8 + wv;
    const long NROWS = (long)BATCH * HEADS * SEQ;   // 262144 per tensor

    float* buf; const float* w; long r;
    if (row < NROWS) { buf = qb; w = qw; r = row; }
    else             { buf = kb; w = kw; r = row - NROWS; }

    float2 x = *(float2*)(&buf[r * DH + lane * 2]);
    float ss = x.x * x.x + x.y * x.y;
    ss += __shfl_xor(ss, 1, 32);
    ss += __shfl_xor(ss, 2, 32);
    ss += __shfl_xor(ss, 4, 32);
    ss += __shfl_xor(ss, 8, 32);
    ss += __shfl_xor(ss, 16, 32);
    const float rinv = rsqrtf(ss * (1.0f / DH) + 1e-6f);
    x.x *= rinv * w[lane * 2];
    x.y *= rinv * w[lane * 2 + 1];
    *(float2*)(&buf[r * DH + lane * 2]) = x;
}

// ---------------------------------------------------------------------------
// Kernel 3: flash attention per (b,h). 128 threads = 4 waves, each wave owns
// 16 query rows; online softmax over 16 key blocks of 64.
// ---------------------------------------------------------------------------
__global__ void __launch_bounds__(128)
attn_kernel(const float* __restrict__ qg, const float* __restrict__ kg,
            const float* __restrict__ vg, float* __restrict__ og)
{
    __shared__ float p_lds[4 * 16 * DH];   // per-wave P staging (16x64)
    const int lane = threadIdx.x & 31;
    const int wv   = threadIdx.x >> 5;
    const int lh   = lane & 15, lg = lane >> 4;
    const int qblk = blockIdx.x;           // 0..15
    const int bh   = blockIdx.y;           // 0..255
    const int bidx = bh >> 4, h = bh & (HEADS - 1);
    const size_t head_base = (size_t)bh * SEQ * DH;
    const int qrow0 = qblk * 64 + wv * 16;

    // Preload scaled Q fragments (A-layout, 16x4 chunks over d)
    v2f qa[16];
    #pragma unroll
    for (int ds = 0; ds < 16; ++ds) {
        v2f t = *(const v2f*)(&qg[head_base + (size_t)(qrow0 + lh) * DH +
                                  ds * 4 + 2 * lg]);
        qa[ds] = t * 0.125f;   // Dh^-0.5
    }

    v8f o0 = {}, o1 = {}, o2 = {}, o3 = {};
    float rmax[8], rsum[8];
    #pragma unroll
    for (int v = 0; v < 8; ++v) { rmax[v] = -3.0e38f; rsum[v] = 0.0f; }

    float* pw = &p_lds[wv * 16 * DH];

    for (int j = 0; j < 16; ++j) {
        const int kr0 = j * 64;

        // Prefetch next key block of K and V (global_prefetch_b8)
        if (j + 1 < 16) {
            __builtin_prefetch(&kg[head_base + (size_t)(kr0 + 64 + (threadIdx.x & 63)) * DH], 0, 1);
            __builtin_prefetch(&vg[head_base + (size_t)(kr0 + 64 + (threadIdx.x & 63)) * DH], 0, 1);
        }

        // --- S = (q*scale) @ K^T for 64 keys: four 16-wide column tiles ---
        v8f s0 = {}, s1 = {}, s2 = {}, s3 = {};
        #pragma unroll
        for (int ds = 0; ds < 16; ++ds) {
            const size_t kbase = head_base + (size_t)(kr0 + lh) * DH +
                                 ds * 4 + 2 * lg;
            v2f b0 = *(const v2f*)(&kg[kbase]);
            v2f b1 = *(const v2f*)(&kg[kbase + 16 * DH]);
            v2f b2 = *(const v2f*)(&kg[kbase + 32 * DH]);
            v2f b3 = *(const v2f*)(&kg[kbase + 48 * DH]);
            s0 = wmma4(qa[ds], b0, s0);
            s1 = wmma4(qa[ds], b1, s1);
            s2 = wmma4(qa[ds], b2, s2);
            s3 = wmma4(qa[ds], b3, s3);
        }

        // --- online softmax; rows of a C-tile live in 16-lane groups ---
        #pragma unroll
        for (int v = 0; v < 8; ++v) {
            float m = rmax[v];
            m = fmaxf(m, s0[v]); m = fmaxf(m, s1[v]);
            m = fmaxf(m, s2[v]); m = fmaxf(m, s3[v]);
            m = fmaxf(m, __shfl_xor(m, 1, 32));
            m = fmaxf(m, __shfl_xor(m, 2, 32));
            m = fmaxf(m, __shfl_xor(m, 4, 32));
            m = fmaxf(m, __shfl_xor(m, 8, 32));
            const float corr = __expf(rmax[v] - m);
            const float p0 = __expf(s0[v] - m);
            const float p1 = __expf(s1[v] - m);
            const float p2 = __expf(s2[v] - m);
            const float p3 = __expf(s3[v] - m);
            float ps = p0 + p1 + p2 + p3;
            ps += __shfl_xor(ps, 1, 32);
            ps += __shfl_xor(ps, 2, 32);
            ps += __shfl_xor(ps, 4, 32);
            ps += __shfl_xor(ps, 8, 32);
            rsum[v] = rsum[v] * corr + ps;
            rmax[v] = m;
            o0[v] *= corr; o1[v] *= corr; o2[v] *= corr; o3[v] *= corr;
            // stage P row-major [m][key] in per-wave LDS (same-wave DS order)
            const int mrow = v + 8 * lg;
            pw[mrow * DH +      lh] = p0;
            pw[mrow * DH + 16 + lh] = p1;
            pw[mrow * DH + 32 + lh] = p2;
            pw[mrow * DH + 48 + lh] = p3;
        }

        // --- O += P @ V over the 64-key block ---
        #pragma unroll
        for (int ks = 0; ks < 16; ++ks) {
            v2f a = *(const v2f*)(&pw[lh * DH + ks * 4 + 2 * lg]);
            const size_t vb0 = head_base +
                               (size_t)(kr0 + ks * 4 + 2 * lg) * DH + lh;
            v2f b0, b1, b2, b3;
            b0.x = vg[vb0];      b0.y = vg[vb0 + DH];
            b1.x = vg[vb0 + 16]; b1.y = vg[vb0 + DH + 16];
            b2.x = vg[vb0 + 32]; b2.y = vg[vb0 + DH + 32];
            b3.x = vg[vb0 + 48]; b3.y = vg[vb0 + DH + 48];
            o0 = wmma4(a, b0, o0);
            o1 = wmma4(a, b1, o1);
            o2 = wmma4(a, b2, o2);
            o3 = wmma4(a, b3, o3);
        }
    }

    // finalize: O /= rowsum, store into [b][n][h*64+d] for the proj GEMM
    #pragma unroll
    for (int v = 0; v < 8; ++v) {
        const float inv = 1.0f / rsum[v];
        const int row = qrow0 + v + 8 * lg;
        float* dst = &og[((size_t)bidx * SEQ + row) * CDIM + h * DH];
        dst[     lh] = o0[v] * inv;
        dst[16 + lh] = o1[v] * inv;
        dst[32 + lh] = o2[v] * inv;
        dst[48 + lh] = o3[v] * inv;
    }
}

// ---------------------------------------------------------------------------
// Kernel 4: out = O @ proj_w^T + proj_b
// ---------------------------------------------------------------------------
__global__ void __launch_bounds__(256)
proj_gemm_kernel(const float* __restrict__ A, const float* __restrict__ W,
                 const float* __restrict__ bias, float* __restrict__ out)
{
    __shared__ float As[2 * 128 * 16];
    __shared__ float Bs[2 * 32 * 16];
    const int lane = threadIdx.x & 31;
    const int wv   = threadIdx.x >> 5;
    const int wm   = wv >> 1, wn = wv & 1;
    const int lh   = lane & 15, lg = lane >> 4;
    const int m0   = blockIdx.y * 128;
    const int n0   = blockIdx.x * 32;

    v8f c0 = {}, c1 = {};
    gemm_mainloop_nt(A, W, CDIM, m0, n0, As, Bs, c0, c1);

    const int   n  = n0 + wn * 16 + lh;
    const float bv = bias[n];
    #pragma unroll
    for (int s2 = 0; s2 < 2; ++s2) {
        v8f c;
        if (s2) c = c1; else c = c0;
        #pragma unroll
        for (int v = 0; v < 8; ++v) {
            const int m = m0 + wm * 32 + s2 * 16 + v + 8 * lg;
            out[(size_t)m * CDIM + n] = c[v] + bv;
        }
    }
}

// ---------------------------------------------------------------------------
extern "C" void kernel_launch(void* const* d_in, const int* in_sizes, int n_in,
                              void* d_out, int out_size, void* d_ws,
                              size_t ws_size, hipStream_t stream)
{
    const float* x      = (const float*)d_in[0];
    const float* qkv_w  = (const float*)d_in[1];
    const float* proj_w = (const float*)d_in[2];
    const float* proj_b = (const float*)d_in[3];
    const float* qnw    = (const float*)d_in[4];
    const float* knw    = (const float*)d_in[5];
    float* out = (float*)d_out;

    const size_t SEG = (size_t)BATCH * HEADS * SEQ * DH;  // 16,777,216 floats
    float* q = (float*)d_ws;
    float* k = q + SEG;
    float* v = k + SEG;
    float* o = v + SEG;

    // 1) QKV GEMM: [16384,1024] x [3072,1024]^T
    qkv_gemm_kernel<<<dim3(3 * CDIM / 32, (BATCH * SEQ) / 128), 256, 0, stream>>>(
        x, qkv_w, q, k, v);

    // 2) RMSNorm on q and k (2 * 262144 rows, 8 rows/block)
    rmsnorm_kernel<<<dim3((2 * BATCH * HEADS * SEQ) / 8), 256, 0, stream>>>(
        q, k, qnw, knw);

    // 3) Attention per (b,h), 64 query rows per block
    attn_kernel<<<dim3(SEQ / 64, BATCH * HEADS), 128, 0, stream>>>(q, k, v, o);

    // 4) Projection GEMM + bias
    proj_gemm_kernel<<<dim3(CDIM / 32, (BATCH * SEQ) / 128), 256, 0, stream>>>(
        o, proj_w, proj_b, out);
}